// MovingPoints_66271345377620
// MI455X (gfx1250) — compile-verified
//
#include <hip/hip_runtime.h>

// MI455X / gfx1250, wave32. H = sum(p^2)/2 + sum_{i!=j} sum_{9 shifts} 1/||xi - xj - s||.
// 16x16 pair tiles; d^2 tile per shift via V_WMMA_F32_16X16X4_F32 on Gram-augmented
// vectors; one v_rsq_f32 per pair-image (the irreducible op). Cancellation-prone
// close pairs are fixed up in a RARE wave-uniform scalar branch (ballot-gated) so
// the hot loop stays WMMA + rsq only. Diagonal tiles (512/131328) take a separate
// exact masked path. Symmetric tile enumeration halves work. Deterministic
// two-stage reduction (no float atomics).

typedef __attribute__((ext_vector_type(2))) float v2f;
typedef __attribute__((ext_vector_type(8))) float v8f;

#define NPTS   8192
#define TILE   16
#define NTILES (NPTS / TILE)   // 512
#define TSHIFT 9               // log2(NTILES)
#define BLOCK  256
#define WPB    (BLOCK / 32)
#define GRID_BLOCKS 4096
#define BOX    10.0f
#define FIX_THRESH 0.01f       // d < 0.1: Gram cancellation may matter -> exact recompute

__device__ __forceinline__ float rsq(float x) { return __builtin_amdgcn_rsqf(x); }

__global__ void pe_tiles_kernel(const float* __restrict__ xy,
                                float* __restrict__ partials) {
    const int lane = threadIdx.x & 31;
    const int lo   = lane & 15;   // column N (and load index)
    const int hi   = lane >> 4;   // half-wave select
    const int wave = blockIdx.x * WPB + (threadIdx.x >> 5);
    const int nwaves = gridDim.x * WPB;

    float acc0 = 0.0f, acc1 = 0.0f;

    const int ntasks = NTILES * NTILES;
    for (int task = wave; task < ntasks; task += nwaves) {
        const int ti = task >> TSHIFT;
        const int tj = task & (NTILES - 1);
        if (ti > tj) continue;                 // symmetry: upper triangle only
        const int bi = ti * TILE, bj = tj * TILE;

        const float2 pi = ((const float2*)xy)[bi + lo];
        const float xi = pi.x, yi = pi.y;
        const float2 pj = ((const float2*)xy)[bj + lo];
        const float xj = pj.x, yj = pj.y;

        // Shifted column coords + squares, hoisted once per tile.
        float xsv[3], ysv[3], xs2v[3], ys2v[3];
#pragma unroll
        for (int k = 0; k < 3; ++k) {
            xsv[k] = xj + (float)(k - 1) * BOX;  xs2v[k] = xsv[k] * xsv[k];
            ysv[k] = yj + (float)(k - 1) * BOX;  ys2v[k] = ysv[k] * ysv[k];
        }

        if (ti != tj) {
            // ---- hot path: WMMA Gram distances, rare exact fixup ----
            const float r2i = xi * xi + yi * yi;
            v2f a;                              // row M = [-2x, -2y, x^2+y^2, 1]
            a.x = hi ? r2i  : (-2.0f * xi);
            a.y = hi ? 1.0f : (-2.0f * yi);

            float t0 = 0.0f, t1 = 0.0f;
#pragma unroll
            for (int si = 0; si < 3; ++si) {
#pragma unroll
                for (int sj = 0; sj < 3; ++sj) {
                    const float xs = xsv[si], ys = ysv[sj];
                    v2f b;                      // col N = [xs, ys, 1, xs^2+ys^2]
                    b.x = hi ? 1.0f : xs;
                    b.y = hi ? (xs2v[si] + ys2v[sj]) : ys;

                    v8f c = {};
                    c = __builtin_amdgcn_wmma_f32_16x16x4_f32(
                            false, a, false, b, (short)0, c, false, false);

                    // Wave-uniform rare branch: only if some element is close.
                    const float mn = fminf(fminf(fminf(c[0], c[1]), fminf(c[2], c[3])),
                                           fminf(fminf(c[4], c[5]), fminf(c[6], c[7])));
                    if (__builtin_amdgcn_ballot_w32(mn < FIX_THRESH)) {
#pragma unroll
                        for (int r = 0; r < 8; ++r) {
                            const float xm = __shfl(xi, r + 8 * hi, 32);
                            const float ym = __shfl(yi, r + 8 * hi, 32);
                            const float ddx = xm - xs, ddy = ym - ys;
                            const float e = ddx * ddx + ddy * ddy;
                            if (c[r] < FIX_THRESH) c[r] = e;   // exact recompute
                        }
                    }
#pragma unroll
                    for (int r = 0; r < 8; r += 2) {
                        t0 += rsq(c[r]);
                        t1 += rsq(c[r + 1]);
                    }
                }
            }
            acc0 += 2.0f * (t0 + t1);           // off-diagonal tiles counted twice
        } else {
            // ---- diagonal tile: exact VALU path with i==j mask (rare) ----
            float xim[8], yim[8];
#pragma unroll
            for (int r = 0; r < 8; ++r) {
                xim[r] = __shfl(xi, r + 8 * hi, 32);
                yim[r] = __shfl(yi, r + 8 * hi, 32);
            }
            float ts = 0.0f;
#pragma unroll
            for (int si = 0; si < 3; ++si) {
#pragma unroll
                for (int sj = 0; sj < 3; ++sj) {
                    const float xs = xsv[si], ys = ysv[sj];
#pragma unroll
                    for (int r = 0; r < 8; ++r) {
                        const float ddx = xim[r] - xs, ddy = yim[r] - ys;
                        const float d2 = ddx * ddx + ddy * ddy;
                        const bool masked = (r + 8 * hi) == lo;  // global i == j
                        ts += masked ? 0.0f : rsq(d2);
                    }
                }
            }
            acc1 += ts;
        }
    }

    // Deterministic in-block tree reduction.
    __shared__ float red[BLOCK];
    red[threadIdx.x] = acc0 + acc1;
    __syncthreads();
    for (int s = BLOCK / 2; s > 0; s >>= 1) {
        if (threadIdx.x < s) red[threadIdx.x] += red[threadIdx.x + s];
        __syncthreads();
    }
    if (threadIdx.x == 0) partials[blockIdx.x] = red[0];
}

__global__ void finalize_kernel(const float* __restrict__ pxy,
                                const float* __restrict__ partials,
                                int nPartials,
                                float* __restrict__ out) {
    float s = 0.0f;
    for (int i = threadIdx.x; i < nPartials; i += BLOCK) s += partials[i];
    for (int i = threadIdx.x; i < 2 * NPTS; i += BLOCK) {
        const float p = pxy[i];
        s += 0.5f * p * p;                      // KE: p^2 / (2*MASS), MASS == 1
    }
    __shared__ float red[BLOCK];
    red[threadIdx.x] = s;
    __syncthreads();
    for (int st = BLOCK / 2; st > 0; st >>= 1) {
        if (threadIdx.x < st) red[threadIdx.x] += red[threadIdx.x + st];
        __syncthreads();
    }
    if (threadIdx.x == 0) out[0] = red[0];
}

extern "C" void kernel_launch(void* const* d_in, const int* in_sizes, int n_in,
                              void* d_out, int out_size, void* d_ws, size_t ws_size,
                              hipStream_t stream) {
    const float* xy  = (const float*)d_in[0];   // [8192, 2] float32
    const float* pxy = (const float*)d_in[1];   // [8192, 2] float32
    float* out       = (float*)d_out;           // scalar float32
    float* partials  = (float*)d_ws;

    int grid = GRID_BLOCKS;
    if ((size_t)grid * sizeof(float) > ws_size) grid = (int)(ws_size / sizeof(float));
    if (grid < 1) grid = 1;

    pe_tiles_kernel<<<grid, BLOCK, 0, stream>>>(xy, partials);
    finalize_kernel<<<1, BLOCK, 0, stream>>>(pxy, partials, grid, out);
}